// RWTransformerForNodeClassification_40587440947424
// MI455X (gfx1250) — compile-verified
//
#include <hip/hip_runtime.h>
#include <stdint.h>

// ---------------- problem constants (match reference) ----------------
#define NUM_NODES   40000
#define DEG         32
#define NUM_EDGES   (NUM_NODES * DEG)      // 1,280,000
#define N_START     4096
#define NUM_WALKS   4
#define WALK_LEN    64
#define S_TOTAL     (NUM_WALKS * WALK_LEN) // 256
#define WINDOW      8
#define D_FEAT      128
#define D_OUT       (D_FEAT + (2 * WINDOW - 1) + 1)   // 144
#define TOTAL_CHAINS (N_START * NUM_WALKS)            // 16384

typedef __attribute__((ext_vector_type(2))) float v2f;
typedef __attribute__((ext_vector_type(8))) float v8f;

#if defined(__gfx1250__) && __has_builtin(__builtin_amdgcn_wmma_f32_16x16x4_f32)
#define HAVE_WMMA_F32_16X16X4 1
#endif

#if defined(__gfx1250__)
  #if __has_builtin(__builtin_amdgcn_global_load_async_to_lds_b128)
    #define HAVE_ASYNC_LDS 1
  #endif
  #if __has_builtin(__builtin_amdgcn_s_wait_asynccnt)
    #define WAIT_ASYNC() __builtin_amdgcn_s_wait_asynccnt(0)
  #else
    #define WAIT_ASYNC() asm volatile("s_wait_asynccnt 0" ::: "memory")
  #endif
#endif

#if defined(HAVE_ASYNC_LDS)
// The builtin's params (from the round-2 diagnostic) are pointers to a
// 16-byte GCC vector of int, with AS(1) (printed as __device__) on the
// global side and AS(3) on the LDS side.
typedef int v4i_mem __attribute__((vector_size(16)));
typedef __attribute__((address_space(3))) v4i_mem lds_v4i;
typedef __attribute__((address_space(1))) v4i_mem glb_v4i;

// Flat->LDS generic pointers truncate to the low 32 bits (ISA aperture rule),
// so building the AS(3) pointer from the low dword is exact.
__device__ __forceinline__ lds_v4i* to_lds(void* p) {
    return (lds_v4i*)(uint32_t)(uintptr_t)p;
}
__device__ __forceinline__ glb_v4i* to_glb(const void* p) {
    return (glb_v4i*)(uintptr_t)p;
}
#endif

// ---------------- Threefry-2x32 (20 rounds) ----------------
__device__ __forceinline__ uint32_t rotl32(uint32_t x, uint32_t r) {
    return (x << r) | (x >> (32u - r));
}

__device__ __forceinline__ void threefry2x32(uint32_t k0, uint32_t k1,
                                             uint32_t c0, uint32_t c1,
                                             uint32_t& o0, uint32_t& o1) {
    constexpr uint32_t R[8] = {13u, 15u, 26u, 6u, 17u, 29u, 16u, 24u};
    uint32_t ks[3] = {k0, k1, k0 ^ k1 ^ 0x1BD11BDAu};
    uint32_t x0 = c0 + ks[0];
    uint32_t x1 = c1 + ks[1];
#pragma unroll
    for (int grp = 0; grp < 5; ++grp) {
        const int base = (grp & 1) ? 4 : 0;
#pragma unroll
        for (int rr = 0; rr < 4; ++rr) {
            x0 += x1;
            x1 = rotl32(x1, R[base + rr]);
            x1 ^= x0;
        }
        x0 += ks[(grp + 1) % 3];
        x1 += ks[(grp + 2) % 3] + (uint32_t)(grp + 1);
    }
    o0 = x0;
    o1 = x1;
}

__device__ __forceinline__ float uniform01(uint32_t r) {
    return ((float)(r >> 8) + 0.5f) * 0x1p-24f;   // (0,1)
}

__device__ __forceinline__ float gumbel_from(uint32_t r) {
    float u = uniform01(r);
    return -__logf(-__logf(u));
}

// =====================================================================
// Kernel 1: per-node sorted hash blocks (wave32 bitonic sort in lanes).
// =====================================================================
__global__ void build_hashes_kernel(const int* __restrict__ col,
                                    int* __restrict__ hashes) {
    const int wave = (int)((blockIdx.x * blockDim.x + threadIdx.x) >> 5);
    const int lane = (int)(threadIdx.x & 31u);
    if (wave >= NUM_NODES) return;

    int c = col[wave * DEG + lane];
#pragma unroll
    for (int k = 2; k <= 32; k <<= 1) {
#pragma unroll
        for (int j = k >> 1; j > 0; j >>= 1) {
            int partner = __shfl_xor(c, j, 32);
            bool dirUp  = ((lane & k) == 0);
            bool lower  = ((lane & j) == 0);
            int mn = min(c, partner), mx = max(c, partner);
            c = (lower == dirUp) ? mn : mx;
        }
    }
    hashes[wave * DEG + lane] = wave * NUM_NODES + c;
}

// =====================================================================
// Kernel 2: node2vec biased walks (one wave32 per chain; lane = neighbor;
// shuffle-binary-search adjacency; Gumbel-max argmax via shfl_xor).
// =====================================================================
__global__ void walk_kernel(const int* __restrict__ col,
                            const int* __restrict__ hashes,
                            const int* __restrict__ start_nodes,
                            int* __restrict__ walks) {
    const int wave = (int)((blockIdx.x * blockDim.x + threadIdx.x) >> 5);
    const int lane = (int)(threadIdx.x & 31u);
    if (wave >= TOTAL_CHAINS) return;
    const uint32_t chain = (uint32_t)wave;

    int cur = start_nodes[wave / NUM_WALKS];
    if (lane == 0) walks[wave * WALK_LEN + 0] = cur;

    uint32_t r0, r1;
    threefry2x32(42u, 0u, chain, 0u, r0, r1);
    int off = (int)(uniform01(r0) * (float)DEG);
    off = min(off, DEG - 1);
    int prev = cur;
    cur = col[prev * DEG + off];
    if (lane == 0) walks[wave * WALK_LEN + 1] = cur;

    for (int step = 2; step < WALK_LEN; ++step) {
        int nbr  = col[cur * DEG + lane];           // coalesced 128B
        int elem = hashes[prev * DEG + lane];       // sorted block of prev, in lanes
        int q    = prev * NUM_NODES + nbr;          // fits int32

        int lo = 0;                                  // lower_bound via shuffles
#pragma unroll
        for (int sh = 16; sh >= 1; sh >>= 1) {
            int probe = __shfl(elem, lo + sh - 1, 32);
            if (probe < q) lo += sh;
        }
        int atlo = __shfl(elem, min(lo, 31), 32);
        bool is_adj = (lo < 32) && (atlo == q);
        bool is_ret = (nbr == prev);

        float w = is_ret ? 0.25f : (is_adj ? 1.0f : 2.0f);   // 1/P, 1, 1/Q
        threefry2x32(42u, (uint32_t)step, chain, (uint32_t)lane, r0, r1);
        float score = __logf(w) + gumbel_from(r0);

        int idx = lane;                              // argmax, lowest lane on tie
#pragma unroll
        for (int d = 16; d >= 1; d >>= 1) {
            float os = __shfl_xor(score, d, 32);
            int   oi = __shfl_xor(idx, d, 32);
            if (os > score || (os == score && oi < idx)) { score = os; idx = oi; }
        }
        int chosen = __shfl(nbr, idx, 32);

        prev = cur;
        cur  = chosen;
        if (lane == 0) walks[wave * WALK_LEN + step] = cur;
    }
}

// =====================================================================
// Kernel 3: output assembly. Hash blocks staged to LDS via
// global_load_async_to_lds_b128 (+ s_wait_asynccnt), anon via LDS scan,
// PE via LDS-local binary search, x-gather via chained
// V_WMMA_F32_16X16X4_F32 identity multiplies (B fragments preloaded).
// =====================================================================
__global__ void assemble_kernel(const float* __restrict__ x,
                                const int* __restrict__ hashes,
                                const int* __restrict__ walks,
                                float* __restrict__ out) {
    __shared__ int s_nodes[S_TOTAL];
    __shared__ int s_hash[S_TOTAL * DEG];   // 32 KB

    const int b = (int)blockIdx.x;
    const int t = (int)threadIdx.x;         // 0..255 walk-order position

    const int node = walks[b * S_TOTAL + t];
    s_nodes[t] = node;

    __builtin_prefetch(x + (size_t)node * D_FEAT, 0, 1);   // global_prefetch_b8

    const int* blk = hashes + node * DEG;   // 128B aligned

#if defined(HAVE_ASYNC_LDS)
    // 8 x async B128: MEM -> LDS, tracked by ASYNCcnt (no VGPR round trip)
#pragma unroll
    for (int i = 0; i < DEG; i += 4) {
        __builtin_amdgcn_global_load_async_to_lds_b128(
            to_glb(blk + i), to_lds(&s_hash[t * DEG + i]),
            /*offset=*/0, /*cpol=*/0);
    }
    WAIT_ASYNC();
#else
#pragma unroll
    for (int i = 0; i < DEG; i += 4) {
        int4 v = *(const int4*)(blk + i);
        *(int4*)(&s_hash[t * DEG + i]) = v;
    }
#endif
    __syncthreads();

    // ---- anon: first occurrence index in walk order ----
    int first = t;
    for (int j = 0; j < S_TOTAL; ++j) {
        int vj = s_nodes[j];
        if (vj == node && j < first) first = j;
    }

    // ---- PE flags (walk-local, l = t & 63) ----
    const int l = t & (WALK_LEN - 1);
    float pe[2 * WINDOW - 1];
#pragma unroll
    for (int k = 1; k <= WINDOW; ++k)
        pe[k - 1] = (l >= k && s_nodes[t - k] == node) ? 1.0f : 0.0f;
#pragma unroll
    for (int k = 2; k <= WINDOW; ++k) {
        float c = 0.0f;
        if (l >= k) {
            int q = node * NUM_NODES + s_nodes[t - k];
            int lo = 0;
#pragma unroll
            for (int sh = 16; sh >= 1; sh >>= 1) {
                if (s_hash[t * DEG + lo + sh - 1] < q) lo += sh;
            }
            if (lo < DEG && s_hash[t * DEG + lo] == q) c = 1.0f;
        }
        pe[WINDOW + (k - 2)] = c;
    }

    const int srow = S_TOTAL - 1 - t;        // flipped output position
    float* orow = out + ((size_t)b * S_TOTAL + srow) * D_OUT;
    {
        // pe region starts at byte 512 of a 576B row -> 16B aligned
        float4* pdst = (float4*)(orow + D_FEAT);
        pdst[0] = make_float4(pe[0],  pe[1],  pe[2],  pe[3]);
        pdst[1] = make_float4(pe[4],  pe[5],  pe[6],  pe[7]);
        pdst[2] = make_float4(pe[8],  pe[9],  pe[10], pe[11]);
        pdst[3] = make_float4(pe[12], pe[13], pe[14], (float)first);
    }

    // ---- x gather: D = I16 x X_tile via 4 chained WMMA f32 16x16x4 ----
#if defined(HAVE_WMMA_F32_16X16X4)
    {
        const int wave = t >> 5;             // 0..7
        const int lane = t & 31;
        const int row  = lane & 15;          // N index for B/D, M index for A
        const int half = lane >> 4;          // VGPR lane-half

        // A identity fragments: constant per lane across all tiles
        v2f Af[4];
#pragma unroll
        for (int c4 = 0; c4 < 4; ++c4) {
            const int k0 = 4 * c4 + 2 * half;
            Af[c4].x = (row == k0)     ? 1.0f : 0.0f;
            Af[c4].y = (row == k0 + 1) ? 1.0f : 0.0f;
        }

        for (int tg = wave; tg < 16; tg += 8) {        // 16 row-tiles of 16 positions
            // per-tile node ids (constant across fg)
            const float* xb0[4];
            const float* xb1[4];
#pragma unroll
            for (int c4 = 0; c4 < 4; ++c4) {
                const int k0 = 4 * c4 + 2 * half;
                xb0[c4] = x + (size_t)s_nodes[tg * 16 + k0]     * D_FEAT;
                xb1[c4] = x + (size_t)s_nodes[tg * 16 + k0 + 1] * D_FEAT;
            }
            for (int fg = 0; fg < 8; ++fg) {           // 8 feature-tiles of 16
                // preload all 8 B elements so loads batch into one clause
                v2f Bf[4];
#pragma unroll
                for (int c4 = 0; c4 < 4; ++c4) {
                    Bf[c4].x = xb0[c4][fg * 16 + row];
                    Bf[c4].y = xb1[c4][fg * 16 + row];
                }
                v8f acc = {};
#pragma unroll
                for (int c4 = 0; c4 < 4; ++c4) {
                    acc = __builtin_amdgcn_wmma_f32_16x16x4_f32(
                        false, Af[c4], false, Bf[c4],
                        (short)0, acc, false, false);
                }
                // C/D layout: lane L, vgpr v -> M = v + 8*(L>=16), N = L&15
#pragma unroll
                for (int v = 0; v < 8; ++v) {
                    const int m  = v + 8 * half;
                    const int so = S_TOTAL - 1 - (tg * 16 + m);
                    out[((size_t)b * S_TOTAL + so) * D_OUT + fg * 16 + row] = acc[v];
                }
            }
        }
    }
#else
    {
        const float* xr = x + (size_t)node * D_FEAT;
        for (int i = 0; i < D_FEAT; i += 4) {
            float4 v = *(const float4*)(xr + i);
            orow[i + 0] = v.x; orow[i + 1] = v.y;
            orow[i + 2] = v.z; orow[i + 3] = v.w;
        }
    }
#endif
}

// =====================================================================
// Host-side launcher
// =====================================================================
extern "C" void kernel_launch(void* const* d_in, const int* in_sizes, int n_in,
                              void* d_out, int out_size, void* d_ws, size_t ws_size,
                              hipStream_t stream) {
    (void)in_sizes; (void)n_in; (void)out_size; (void)ws_size;

    const float* x           = (const float*)d_in[0];
    const int*   edge_index  = (const int*)d_in[1];     // [2, E] flat
    const int*   start_nodes = (const int*)d_in[2];
    const int*   col         = edge_index + NUM_EDGES;  // row half implied by layout

    int* hashes = (int*)d_ws;                           // E int32 (5.12 MB)
    int* walks  = hashes + NUM_EDGES;                   // 16384*64 int32 (4 MB)
    float* out  = (float*)d_out;

    build_hashes_kernel<<<(NUM_NODES + 3) / 4, 128, 0, stream>>>(col, hashes);
    walk_kernel<<<TOTAL_CHAINS / 4, 128, 0, stream>>>(col, hashes, start_nodes, walks);
    assemble_kernel<<<N_START, S_TOTAL, 0, stream>>>(x, hashes, walks, out);
}